// Faster_NDCG_v2_11098195493107
// MI455X (gfx1250) — compile-verified
//
#include <hip/hip_runtime.h>

// Faster_NDCG_v2 fused loss for MI455X (gfx1250, wave32).
// Kernel A: one block (256 threads = 8 waves) per batch row b.
//   Stage 1: row load -> LDS, n_fin reduction.
//   Stage 2: pairwise squared hinge row-sums g[i] via V_WMMA_F32_16X16X4_F32
//            (A = 16x4 hinge tile, B = ones -> D rows hold sum_j, C accumulates).
//   Stage 3: per-item terms, three row reductions, per_item row mean.
// Kernel B: loss = (sum_a num_pos[a]/(idcg[a]+EPS)) * (sum_c rowmean[c]) / B^2.

#define SS 512
#define SP2 514          // (S + 2) stride of u/v tables
#define EPSV 1e-10f
#define TAU1V 1e-3f
#define TAU2V 1e-4f
#define GAMMAV 0.1f
#define ETA1V 0.01f
#define LN2V 0.6931471805599453f
#define PADV -1.0f
#define NEGBIG -1e30f

typedef __attribute__((ext_vector_type(2))) float v2f;
typedef __attribute__((ext_vector_type(8))) float v8f;

__device__ __forceinline__ float block_reduce_256(float v, float* scr) {
  const int t = threadIdx.x;
  scr[t] = v;
  __syncthreads();
#pragma unroll
  for (int s = 128; s > 0; s >>= 1) {
    if (t < s) scr[t] += scr[t + s];
    __syncthreads();
  }
  float r = scr[0];
  __syncthreads();
  return r;
}

__launch_bounds__(256)
__global__ void ndcg_row_kernel(const float* __restrict__ y_pred,
                                const float* __restrict__ y_true,
                                const int*   __restrict__ num_item,
                                const float* __restrict__ u_tab,
                                const float* __restrict__ v_tab,
                                const float* __restrict__ lambda_q,
                                const float* __restrict__ r_q,
                                const float* __restrict__ s_q,
                                float* __restrict__ row_mean) {
  __shared__ float lds_yp[SS];    // pad -> 0 (stage-3 values, WMMA i-reads)
  __shared__ float lds_ypg[SS];   // pad -> -1e30 (WMMA j-reads: hinge exactly 0)
  __shared__ float lds_g[SS];
  __shared__ float scr[256];

  const int b    = blockIdx.x;
  const int t    = threadIdx.x;
  const int lane = t & 31;
  const int wave = t >> 5;

  // ---------------- stage 1: load row, masks, n_fin ----------------
  const float p0  = y_pred[b * SS + t];
  const float p1  = y_pred[b * SS + t + 256];
  const float yt0 = y_true[b * SS + t];
  const float yt1 = y_true[b * SS + t + 256];
  const bool pad0 = (yt0 == PADV);
  const bool pad1 = (yt1 == PADV);
  lds_yp[t]        = pad0 ? 0.f : p0;
  lds_yp[t + 256]  = pad1 ? 0.f : p1;
  lds_ypg[t]       = pad0 ? NEGBIG : p0;
  lds_ypg[t + 256] = pad1 ? NEGBIG : p1;

  // warm L2/L0 for the stage-3 u/v row reads (global_prefetch_b8)
  const int uvbase = (b + 1) * SP2 + 1;
  __builtin_prefetch(&u_tab[uvbase + t], 0, 0);
  __builtin_prefetch(&u_tab[uvbase + t + 256], 0, 0);
  __builtin_prefetch(&v_tab[uvbase + t], 0, 0);
  __builtin_prefetch(&v_tab[uvbase + t + 256], 0, 0);

  const float n_fin = block_reduce_256((pad0 ? 0.f : 1.f) + (pad1 ? 0.f : 1.f), scr);
  const float inv_n = 1.0f / n_fin;

  // ---------------- stage 2: g[i] via f32 WMMA row-sum trick ----------------
  // D = A x B + C with A[m,k] = hinge(i0+m, j0+k) (16x4, f32),
  // B = ones(4x16) -> D[m,n] = sum_k A[m,k] for every n; accumulate over j chunks.
  {
    v2f bones;
    bones[0] = 1.0f;
    bones[1] = 1.0f;
    const int jsel = (lane >> 4) << 1;  // lanes 0-15: K j-offsets {0,1}; 16-31: {2,3}
    for (int grp = wave; grp < 32; grp += 8) {
      const int i0 = grp << 4;
      const int m  = lane & 15;
      const float yi1 = 1.0f - lds_yp[i0 + m];   // d+1 = yp_j + (1 - yp_i)
      v8f c0 = {0.f, 0.f, 0.f, 0.f, 0.f, 0.f, 0.f, 0.f};
      v8f c1 = c0;
#pragma unroll 4
      for (int j0 = 0; j0 < SS; j0 += 8) {
        // chunk A (j0..j0+3)
        {
          const float ya = lds_ypg[j0 + jsel];
          const float yb = lds_ypg[j0 + jsel + 1];
          const float d0 = fmaxf(ya + yi1, 0.f);
          const float d1 = fmaxf(yb + yi1, 0.f);
          v2f a;
          a[0] = d0 * d0;
          a[1] = d1 * d1;
          c0 = __builtin_amdgcn_wmma_f32_16x16x4_f32(false, a, false, bones,
                                                     (short)0, c0, false, false);
        }
        // chunk B (j0+4..j0+7)
        {
          const float ya = lds_ypg[j0 + 4 + jsel];
          const float yb = lds_ypg[j0 + 4 + jsel + 1];
          const float d0 = fmaxf(ya + yi1, 0.f);
          const float d1 = fmaxf(yb + yi1, 0.f);
          v2f a;
          a[0] = d0 * d0;
          a[1] = d1 * d1;
          c1 = __builtin_amdgcn_wmma_f32_16x16x4_f32(false, a, false, bones,
                                                     (short)0, c1, false, false);
        }
      }
      const v8f c = c0 + c1;
      // D layout: lanes 0-15 VGPR r -> M=r; lanes 16-31 VGPR r -> M=8+r.
      if (lane == 0) {
#pragma unroll
        for (int r = 0; r < 8; ++r) {
          const int i = i0 + r;
          const float fin_i = (lds_ypg[i] < -1e29f) ? 0.f : 1.f;
          lds_g[i] = fin_i * c[r] * inv_n + EPSV;
        }
      } else if (lane == 16) {
#pragma unroll
        for (int r = 0; r < 8; ++r) {
          const int i = i0 + 8 + r;
          const float fin_i = (lds_ypg[i] < -1e29f) ? 0.f : 1.f;
          lds_g[i] = fin_i * c[r] * inv_n + EPSV;
        }
      }
    }
  }
  __syncthreads();

  // ---------------- stage 3: per-item terms + row reductions ----------------
  const float lam  = lambda_q[b + 1];
  const float s_qv = s_q[b + 1];
  const float r_qv = r_q[b + 1];
  const float ni   = (float)num_item[b];

  float s_tt = 0.f, s_Gl = 0.f, s_tty = 0.f;
  float nfg_[2], w1fgu_[2], gi_[2], ypv_[2];

#pragma unroll
  for (int k = 0; k < 2; ++k) {
    const int i = t + (k << 8);
    const float yt  = (k == 0) ? yt0 : yt1;
    const bool pad  = (yt == PADV);
    const float fin = pad ? 0.f : 1.f;
    const float ypv = lds_yp[i];
    const float G   = pad ? 0.f : (exp2f(fmaxf(yt, 0.f)) - 1.0f);
    const float pld = pad ? 0.f : (ypv - lam);
    const float sig1 = 1.0f / (1.0f + expf(-pld * (1.0f / TAU1V)));
    const float siga = 1.0f / (1.0f + expf(-pld * 2.0f));
    const float temp_term = sig1 * (1.0f - sig1) * (1.0f / TAU1V);
    const float u_old = u_tab[uvbase + i];
    const float v_old = v_tab[uvbase + i];
    const float g_i   = lds_g[i];
    const float v_new = (1.0f - GAMMAV) * v_old + GAMMAV * (u_old - g_i);
    const float g_u   = u_old - ETA1V * v_new;     // u_new read-back
    const float t2    = 2.0f + ni * g_u;
    const float l2t   = log2f(t2);
    const float nfg = (G * ni) / (l2t * l2t * t2 * LN2V) * 2.0f * siga;
    const float w1  = 2.0f * siga * (1.0f - siga);
    const float fgu = -G / l2t;
    s_tt  += fin * temp_term;
    s_Gl  += fin * (G / l2t);
    s_tty += fin * temp_term * ypv;
    nfg_[k]   = nfg;
    w1fgu_[k] = w1 * fgu;
    gi_[k]    = g_i;
    ypv_[k]   = ypv;
  }

  const float s_tt_tot  = block_reduce_256(s_tt, scr);
  const float s_Gl_tot  = block_reduce_256(s_Gl, scr);
  const float s_tty_tot = block_reduce_256(s_tty, scr);

  const float L_hess      = TAU2V + s_tt_tot * inv_n;
  const float nabla_s_phi = L_hess * s_qv + s_Gl_tot * inv_n;
  const float r_new       = (1.0f - GAMMAV) * r_qv + GAMMAV * nabla_s_phi;
  const float s_new       = s_qv - ETA1V * r_new;
  const float hess        = (s_tty_tot * inv_n) / s_new;

  float local = 0.f;
#pragma unroll
  for (int k = 0; k < 2; ++k)
    local += nfg_[k] * gi_[k] + w1fgu_[k] * (ypv_[k] - hess);

  const float rowsum = block_reduce_256(local, scr);
  if (t == 0) row_mean[b] = rowsum * (1.0f / (float)SS);
}

__launch_bounds__(256)
__global__ void ndcg_final_kernel(const float* __restrict__ rm,
                                  const int*   __restrict__ num_pos,
                                  const float* __restrict__ ideal_dcg,
                                  float* __restrict__ out, int nB) {
  __shared__ float s1[256];
  __shared__ float s2[256];
  const int t = threadIdx.x;
  s1[t] = (t < nB) ? rm[t] : 0.f;
  s2[t] = (t < nB) ? ((float)num_pos[t] / (ideal_dcg[t] + EPSV)) : 0.f;
  __syncthreads();
#pragma unroll
  for (int s = 128; s > 0; s >>= 1) {
    if (t < s) { s1[t] += s1[t + s]; s2[t] += s2[t + s]; }
    __syncthreads();
  }
  if (t == 0) out[0] = s1[0] * s2[0] / ((float)nB * (float)nB);
}

extern "C" void kernel_launch(void* const* d_in, const int* in_sizes, int n_in,
                              void* d_out, int out_size, void* d_ws, size_t ws_size,
                              hipStream_t stream) {
  // setup_inputs() order:
  // 0 y_pred [B,S] f32   1 y_true [B,S] f32   2 qid i32   3 indices i32
  // 4 num_pos [B] i32    5 num_item [B] i32   6 ideal_dcg [B] f32
  // 7 u [(QN+2),(S+2)] f32  8 v f32  9 lambda_q f32  10 z_q f32 (unused)
  // 11 r_q f32  12 s_q f32
  const float* y_pred    = (const float*)d_in[0];
  const float* y_true    = (const float*)d_in[1];
  const int*   num_pos   = (const int*)d_in[4];
  const int*   num_item  = (const int*)d_in[5];
  const float* ideal_dcg = (const float*)d_in[6];
  const float* u_tab     = (const float*)d_in[7];
  const float* v_tab     = (const float*)d_in[8];
  const float* lambda_q  = (const float*)d_in[9];
  const float* r_q       = (const float*)d_in[11];
  const float* s_q       = (const float*)d_in[12];
  float* out = (float*)d_out;
  float* rm  = (float*)d_ws;   // B floats of scratch

  const int nB = in_sizes[4];  // 256

  ndcg_row_kernel<<<nB, 256, 0, stream>>>(y_pred, y_true, num_item, u_tab, v_tab,
                                          lambda_q, r_q, s_q, rm);
  ndcg_final_kernel<<<1, 256, 0, stream>>>(rm, num_pos, ideal_dcg, out, nB);
}